// PEASAGEChannel_35639638622842
// MI455X (gfx1250) — compile-verified
//
#include <hip/hip_runtime.h>

#define NODES 100000
#define EDGES 625000
#define CDIM  128   // input/hidden feature dim for every layer's GEMM K
#define ODIM2 64    // final layer output dim

typedef __attribute__((ext_vector_type(2))) float v2f;
typedef __attribute__((ext_vector_type(8))) float v8f;

// ---------------------------------------------------------------------------
// Scatter: one wave per edge. Lane l adds elements l, l+32, l+64, l+96 of the
// 128-float source row into agg[dst] via f32 atomics (L2-resident: agg is
// 51.2MB < 192MB L2). Lane 0 bumps the degree counter.
// ---------------------------------------------------------------------------
__global__ __launch_bounds__(256) void sage_scatter(
    const float* __restrict__ x, const int* __restrict__ src,
    const int* __restrict__ dst, float* __restrict__ agg,
    float* __restrict__ deg, int nedges)
{
    int e    = (int)((blockIdx.x * blockDim.x + threadIdx.x) >> 5);
    int lane = threadIdx.x & 31;
    if (e >= nedges) return;
    int s = src[e];
    int d = dst[e];
    const float* xs = x   + (size_t)s * CDIM;
    float*       ad = agg + (size_t)d * CDIM;
#pragma unroll
    for (int i = 0; i < 4; ++i)
        atomicAdd(ad + lane + 32 * i, xs[lane + 32 * i]);
    if (lane == 0) atomicAdd(deg + d, 1.0f);
}

// ---------------------------------------------------------------------------
// Fused SAGE layer GEMM:  out = relu?( (agg/max(deg,1)) @ Wl^T + bl + x @ Wr^T )
// One wave owns a 16-row block and ALL O columns (NT = O/16 accumulator tiles).
// K-loop uses V_WMMA_F32_16X16X4_F32; the 1/deg scale is fused into the A load,
// bias + relu into the epilogue. 8 waves per 256-thread block.
// ---------------------------------------------------------------------------
template <int O, bool RELU>
__global__ __launch_bounds__(256) void sage_gemm(
    const float* __restrict__ agg, const float* __restrict__ deg,
    const float* __restrict__ xin,
    const float* __restrict__ Wl, const float* __restrict__ bl,
    const float* __restrict__ Wr,
    float* __restrict__ out, int nrows)
{
    constexpr int NT = O / 16;
    int rowblk = blockIdx.x * (blockDim.x >> 5) + (threadIdx.x >> 5);
    int row0   = rowblk * 16;
    if (row0 >= nrows) return;            // uniform per wave: EXEC stays all-1s

    int lane = threadIdx.x & 31;
    int m    = lane & 15;                 // A row / B-C-D column within tile
    int half = lane >> 4;                 // K-subgroup for A/B fragments

    float rdeg = 1.0f / fmaxf(deg[row0 + m], 1.0f);

    v8f acc[NT];
#pragma unroll
    for (int t = 0; t < NT; ++t)
#pragma unroll
        for (int r = 0; r < 8; ++r) acc[t][r] = 0.0f;

    const float* Ar = agg + (size_t)(row0 + m) * CDIM;
    const float* Xr = xin + (size_t)(row0 + m) * CDIM;

    // Pass 1: mean-aggregated neighbors through Wl (scale fused into A load)
    for (int k0 = 0; k0 < CDIM; k0 += 4) {
        int k = k0 + 2 * half;
        v2f a;
        a.x = Ar[k]     * rdeg;
        a.y = Ar[k + 1] * rdeg;
#pragma unroll
        for (int t = 0; t < NT; ++t) {
            const float* w = Wl + (size_t)(t * 16 + m) * CDIM + k;
            v2f b; b.x = w[0]; b.y = w[1];
            acc[t] = __builtin_amdgcn_wmma_f32_16x16x4_f32(
                false, a, false, b, (short)0, acc[t], false, false);
        }
    }
    // Pass 2: root/self features through Wr
    for (int k0 = 0; k0 < CDIM; k0 += 4) {
        int k = k0 + 2 * half;
        v2f a;
        a.x = Xr[k];
        a.y = Xr[k + 1];
#pragma unroll
        for (int t = 0; t < NT; ++t) {
            const float* w = Wr + (size_t)(t * 16 + m) * CDIM + k;
            v2f b; b.x = w[0]; b.y = w[1];
            acc[t] = __builtin_amdgcn_wmma_f32_16x16x4_f32(
                false, a, false, b, (short)0, acc[t], false, false);
        }
    }

    // Epilogue: bias (+ relu), store. D layout: elem r -> row r+8*half, col m.
#pragma unroll
    for (int t = 0; t < NT; ++t) {
        float bias = bl[t * 16 + m];
#pragma unroll
        for (int r = 0; r < 8; ++r) {
            float v = acc[t][r] + bias;
            if (RELU) v = fmaxf(v, 0.0f);
            out[(size_t)(row0 + r + 8 * half) * O + t * 16 + m] = v;
        }
    }
}

// ---------------------------------------------------------------------------
// Host side
// ---------------------------------------------------------------------------
extern "C" void kernel_launch(void* const* d_in, const int* in_sizes, int n_in,
                              void* d_out, int out_size, void* d_ws, size_t ws_size,
                              hipStream_t stream)
{
    const float* x   = (const float*)d_in[0];
    const int*   eil = (const int*)  d_in[1];   // [3][2][EDGES]
    const float* Wl0 = (const float*)d_in[2];
    const float* bl0 = (const float*)d_in[3];
    const float* Wr0 = (const float*)d_in[4];
    const float* Wl1 = (const float*)d_in[5];
    const float* bl1 = (const float*)d_in[6];
    const float* Wr1 = (const float*)d_in[7];
    const float* Wl2 = (const float*)d_in[8];
    const float* bl2 = (const float*)d_in[9];
    const float* Wr2 = (const float*)d_in[10];

    const size_t featBytes = (size_t)NODES * CDIM * sizeof(float);
    char*  ws  = (char*)d_ws;
    float* agg = (float*)ws;                              // 51.2 MB
    float* deg = (float*)(ws + featBytes);                // 0.4 MB
    float* h1  = (float*)(ws + featBytes + (size_t)NODES * sizeof(float));
    float* h2  = (float*)((char*)h1 + featBytes);

    const int scatterBlocks = (EDGES * 32 + 255) / 256;   // 1 wave / edge
    const int gemmBlocks    = (NODES / 16 + 7) / 8;       // 8 row-blocks / block

    // ---- Layer 0: x -> h1 (relu) ----
    {
        const int* src = eil + (size_t)0 * 2 * EDGES;
        const int* dst = src + EDGES;
        hipMemsetAsync(agg, 0, featBytes, stream);
        hipMemsetAsync(deg, 0, (size_t)NODES * sizeof(float), stream);
        sage_scatter<<<scatterBlocks, 256, 0, stream>>>(x, src, dst, agg, deg, EDGES);
        sage_gemm<128, true><<<gemmBlocks, 256, 0, stream>>>(
            agg, deg, x, Wl0, bl0, Wr0, h1, NODES);
    }
    // ---- Layer 1: h1 -> h2 (relu) ----
    {
        const int* src = eil + (size_t)1 * 2 * EDGES;
        const int* dst = src + EDGES;
        hipMemsetAsync(agg, 0, featBytes, stream);
        hipMemsetAsync(deg, 0, (size_t)NODES * sizeof(float), stream);
        sage_scatter<<<scatterBlocks, 256, 0, stream>>>(h1, src, dst, agg, deg, EDGES);
        sage_gemm<128, true><<<gemmBlocks, 256, 0, stream>>>(
            agg, deg, h1, Wl1, bl1, Wr1, h2, NODES);
    }
    // ---- Layer 2: h2 -> d_out (no relu, O=64) ----
    {
        const int* src = eil + (size_t)2 * 2 * EDGES;
        const int* dst = src + EDGES;
        hipMemsetAsync(agg, 0, featBytes, stream);
        hipMemsetAsync(deg, 0, (size_t)NODES * sizeof(float), stream);
        sage_scatter<<<scatterBlocks, 256, 0, stream>>>(h2, src, dst, agg, deg, EDGES);
        sage_gemm<ODIM2, false><<<gemmBlocks, 256, 0, stream>>>(
            agg, deg, h2, Wl2, bl2, Wr2, (float*)d_out, NODES);
    }
}